// ProjectionLayer_48739288875461
// MI455X (gfx1250) — compile-verified
//
#include <hip/hip_runtime.h>
#include <hip/hip_bf16.h>
#include <math.h>

// ---------------------------------------------------------------------------
// Problem constants (match the reference)
// ---------------------------------------------------------------------------
#define IMG_H 256
#define IMG_W 832
#define IMG_HW (IMG_H * IMG_W)
#define NBATCH 32

typedef float v2f __attribute__((ext_vector_type(2)));
typedef float v8f __attribute__((ext_vector_type(8)));

// ---------------------------------------------------------------------------
// 3x3 matmul on the WMMA unit (V_WMMA_F32_16X16X4_F32, wave32).
// All lanes hold X and Y uniformly (row-major, 9 floats each).
// ISA 7.12.2 layouts:
//   A (16x4 f32):  lane m (0..15): v0=K0, v1=K1 ; lane m+16: v0=K2, v1=K3
//   B (4x16 f32):  v0 lanes0-15 = K0 (col=lane), lanes16-31 = K2
//                  v1 lanes0-15 = K1,            lanes16-31 = K3
//   D (16x16 f32): VGPR v: lanes0-15 row v, col=lane; lanes16-31 row v+8
// We use the 3x3 corner; everything else is zero.
// ---------------------------------------------------------------------------
__device__ __forceinline__ void mm3_wmma(const float* X, const float* Y, float* O) {
    const int  lane = (int)(threadIdx.x & 31u);
    const int  r    = lane & 15;
    const bool hi   = lane >= 16;
    const bool rok  = r < 3;

    v2f A, B;
    A[0] = rok ? (hi ? X[r * 3 + 2] : X[r * 3 + 0]) : 0.0f;   // K=2 : K=0
    A[1] = (rok && !hi) ? X[r * 3 + 1] : 0.0f;                // K=1 (K=3 -> 0)
    B[0] = rok ? (hi ? Y[6 + r] : Y[r]) : 0.0f;               // K=2 row : K=0 row
    B[1] = (rok && !hi) ? Y[3 + r] : 0.0f;                    // K=1 row (K=3 -> 0)

    v8f C = {0.f, 0.f, 0.f, 0.f, 0.f, 0.f, 0.f, 0.f};
    v8f D = __builtin_amdgcn_wmma_f32_16x16x4_f32(
        /*neg_a=*/false, A, /*neg_b=*/false, B,
        /*c_mod=*/(short)0, C, /*reuse_a=*/false, /*reuse_b=*/false);

    // Extract 3x3 corner: D[m][n] lives in VGPR m (m<8), lane n (n<16).
    O[0] = __shfl(D[0], 0, 32); O[1] = __shfl(D[0], 1, 32); O[2] = __shfl(D[0], 2, 32);
    O[3] = __shfl(D[1], 0, 32); O[4] = __shfl(D[1], 1, 32); O[5] = __shfl(D[1], 2, 32);
    O[6] = __shfl(D[2], 0, 32); O[7] = __shfl(D[2], 1, 32); O[8] = __shfl(D[2], 2, 32);
}

// ---------------------------------------------------------------------------
// Per-batch precompute: M = K * R * K^-1 (3x3) and tv = K * t (3), where
// R = xmat @ ymat @ zmat from pose*0.01 (reference _euler2mat order).
// One wave (32 lanes) per batch so WMMA sees EXEC all-ones.
// Output: d_ws[b*12 .. b*12+11] = {M row-major (9), tv (3)}.
// ---------------------------------------------------------------------------
__global__ __launch_bounds__(32) void precompute_proj_kernel(
    const float* __restrict__ pose, const float* __restrict__ K,
    const float* __restrict__ Kinv, float* __restrict__ Mt) {
    const int b = blockIdx.x;

    float p[6];
#pragma unroll
    for (int i = 0; i < 6; ++i) p[i] = pose[b * 6 + i] * 0.01f;
    const float rx = p[3], ry = p[4], rz = p[5];

    const float cz = cosf(rz), sz = sinf(rz);
    const float cy = cosf(ry), sy = sinf(ry);
    const float cx = cosf(rx), sx = sinf(rx);

    const float Z[9] = {cz, -sz, 0.f, sz, cz, 0.f, 0.f, 0.f, 1.f};
    const float Y[9] = {cy, 0.f, sy, 0.f, 1.f, 0.f, -sy, 0.f, cy};
    const float X[9] = {1.f, 0.f, 0.f, 0.f, cx, -sx, 0.f, sx, cx};

    float Km[9], Ki[9];
#pragma unroll
    for (int i = 0; i < 9; ++i) { Km[i] = K[i]; Ki[i] = Kinv[i]; }

    float T1[9], R[9], P3[9], M[9];
    mm3_wmma(Y, Z, T1);   // ymat @ zmat
    mm3_wmma(X, T1, R);   // xmat @ (ymat @ zmat)
    mm3_wmma(Km, R, P3);  // K @ R
    mm3_wmma(P3, Ki, M);  // (K @ R) @ K^-1

    if ((threadIdx.x & 31u) == 0u) {
        float* o = Mt + b * 12;
#pragma unroll
        for (int i = 0; i < 9; ++i) o[i] = M[i];
        o[9]  = Km[0] * p[0] + Km[1] * p[1] + Km[2] * p[2];
        o[10] = Km[3] * p[0] + Km[4] * p[1] + Km[5] * p[2];
        o[11] = Km[6] * p[0] + Km[7] * p[1] + Km[8] * p[2];
    }
}

// ---------------------------------------------------------------------------
// Main warp kernel: memory-bound. 4 consecutive pixels / thread:
//   depth  -> one global_load_b128
//   output -> three global_store_b128 (48B contiguous, 16B aligned)
// Per pixel: unnorm = (M @ [x,y,1]) * depth + tv ; bilinear gather (clamped),
// exactly mirroring the reference (weights from clamped coords).
// ---------------------------------------------------------------------------
__global__ __launch_bounds__(256) void warp_bilinear_kernel(
    const float* __restrict__ img, const float* __restrict__ depth,
    const float* __restrict__ Mt, float* __restrict__ out) {
    const int b  = blockIdx.y;
    const int p0 = (blockIdx.x * 256 + threadIdx.x) * 4;  // grid sized exactly

    const float* mt = Mt + b * 12;
    const float m00 = mt[0], m01 = mt[1], m02 = mt[2];
    const float m10 = mt[3], m11 = mt[4], m12 = mt[5];
    const float m20 = mt[6], m21 = mt[7], m22 = mt[8];
    const float t0 = mt[9], t1 = mt[10], t2 = mt[11];

    const int y  = p0 / IMG_W;
    const int xb = p0 - y * IMG_W;  // 4-px chunk never crosses a row (W%4==0)

    const size_t dbase = (size_t)b * IMG_HW;
    const float4 dv = *(const float4*)(depth + dbase + p0);
    // Prefetch the depth this block will touch ~1 block ahead (speculative-safe).
    __builtin_prefetch(depth + dbase + p0 + 1024, 0, 1);

    const float* __restrict__ src = img + dbase * 3;
    const float  fy = (float)y;

    float res[12];
#pragma unroll
    for (int i = 0; i < 4; ++i) {
        const float fx = (float)(xb + i);
        const float d  = (&dv.x)[i];

        const float u = fmaf(m00, fx, fmaf(m01, fy, m02));
        const float v = fmaf(m10, fx, fmaf(m11, fy, m12));
        const float w = fmaf(m20, fx, fmaf(m21, fy, m22));

        const float un0 = fmaf(u, d, t0);
        const float un1 = fmaf(v, d, t1);
        const float un2 = fmaf(w, d, t2);

        const float zz = un2 + 1e-10f;
        const float px = un0 / zz;
        const float py = un1 / zz;

        const float x0f = floorf(px), y0f = floorf(py);
        const float x1f = x0f + 1.0f, y1f = y0f + 1.0f;

        const float x0s = fminf(fmaxf(x0f, 0.0f), (float)(IMG_W - 1));
        const float x1s = fminf(fmaxf(x1f, 0.0f), (float)(IMG_W - 1));
        const float y0s = fminf(fmaxf(y0f, 0.0f), (float)(IMG_H - 1));
        const float y1s = fminf(fmaxf(y1f, 0.0f), (float)(IMG_H - 1));

        const float wx0 = x1s - px, wx1 = px - x0s;
        const float wy0 = y1s - py, wy1 = py - y0s;

        const int ix0 = (int)x0s, ix1 = (int)x1s;
        const int iy0 = (int)y0s, iy1 = (int)y1s;

        const float* g00 = src + (iy0 * IMG_W + ix0) * 3;  // (y0, x0)
        const float* g10 = src + (iy1 * IMG_W + ix0) * 3;  // (y1, x0)
        const float* g01 = src + (iy0 * IMG_W + ix1) * 3;  // (y0, x1)
        const float* g11 = src + (iy1 * IMG_W + ix1) * 3;  // (y1, x1)

        const float w00 = wx0 * wy0, w01 = wx0 * wy1;
        const float w10 = wx1 * wy0, w11 = wx1 * wy1;

#pragma unroll
        for (int c = 0; c < 3; ++c) {
            res[i * 3 + c] =
                w00 * g00[c] + w01 * g10[c] + w10 * g01[c] + w11 * g11[c];
        }
    }

    float4* op = (float4*)(out + (dbase + (size_t)p0) * 3);
    op[0] = make_float4(res[0], res[1], res[2],  res[3]);
    op[1] = make_float4(res[4], res[5], res[6],  res[7]);
    op[2] = make_float4(res[8], res[9], res[10], res[11]);
}

// ---------------------------------------------------------------------------
// Launcher
//   d_in[0] source_img (32,256,832,3) f32
//   d_in[1] depth_map  (32,256,832,1) f32
//   d_in[2] pose       (32,6)         f32
//   d_in[3] K          (3,3)          f32
//   d_in[4] K_inv      (3,3)          f32
//   d_out   (32,256,832,3) f32
//   d_ws: 32*12 floats of per-batch {M, K*t}
// ---------------------------------------------------------------------------
extern "C" void kernel_launch(void* const* d_in, const int* in_sizes, int n_in,
                              void* d_out, int out_size, void* d_ws, size_t ws_size,
                              hipStream_t stream) {
    const float* img   = (const float*)d_in[0];
    const float* depth = (const float*)d_in[1];
    const float* pose  = (const float*)d_in[2];
    const float* K     = (const float*)d_in[3];
    const float* Kinv  = (const float*)d_in[4];
    float*       out   = (float*)d_out;
    float*       Mt    = (float*)d_ws;  // 32*12 floats

    precompute_proj_kernel<<<dim3(NBATCH), dim3(32), 0, stream>>>(pose, K, Kinv, Mt);

    const int blocks_x = IMG_HW / (256 * 4);  // 208, exact
    warp_bilinear_kernel<<<dim3(blocks_x, NBATCH), dim3(256), 0, stream>>>(
        img, depth, Mt, out);
}